// SparseConvolutionBase_11149735100622
// MI455X (gfx1250) — compile-verified
//
#include <hip/hip_runtime.h>

typedef __attribute__((ext_vector_type(2))) float v2f;
typedef __attribute__((ext_vector_type(4))) float v4f;
typedef __attribute__((ext_vector_type(8))) float v8f;

#define N_IN_V   131072
#define N_OUT_V  131072
#define KVOL     27
#define PAIRS    65536
#define CIN      64
#define COUT     64
#define TILES_PER_K     (PAIRS / 16)            // 4096 tiles of 16 pairs
#define WAVES_PER_BLOCK 8
#define BLOCKS_PER_K    (TILES_PER_K / WAVES_PER_BLOCK)  // 512 -> each block sees one k

// ---------------------------------------------------------------------------
// Init: out[row][c] = bias[c]  (reference: zeros -> scatter-add -> +bias)
// ---------------------------------------------------------------------------
__global__ __launch_bounds__(256)
void sconv_init_bias(const float* __restrict__ bias, float* __restrict__ out) {
    int idx = blockIdx.x * 256 + threadIdx.x;   // grid sized exactly N_OUT*COUT
    out[idx] = bias[idx & (COUT - 1)];
}

// ---------------------------------------------------------------------------
// Main: per-wave 16-pair tile: gather A (16x64) via b128 loads, B=kernel[k]
// from LDS (interleaved pairs -> ds_load_b64 fragments), 64 x
// V_WMMA_F32_16X16X4_F32, scatter-add via global_atomic_add_f32.
//
// K-slot permutation (identical for A and B, so the dot product is unchanged):
//   step 2t   : slots {K0,K1 | K2,K3} = cols {8t+0,8t+1 | 8t+4,8t+5}
//   step 2t+1 : slots {K0,K1 | K2,K3} = cols {8t+2,8t+3 | 8t+6,8t+7}
// => lane half h reads float4 at col 8t + 4h once per step-pair.
// ---------------------------------------------------------------------------
__global__ __launch_bounds__(256)
void sconv_wmma_f32(const float* __restrict__ feats,
                    const float* __restrict__ kern,
                    const int*   __restrict__ in_map,
                    const int*   __restrict__ out_map,
                    float*       __restrict__ out) {
    // kernel[k] staged interleaved: lds[(c>>1)*128 + 2*o + (c&1)] = kern[k][c][o]
    // so the B pair (c even, c+1) at column o is one aligned float2.
    __shared__ float lds_b[CIN * COUT];

    const int tid = threadIdx.x;
    const int k   = blockIdx.x / BLOCKS_PER_K;      // uniform per block

    const float* kb = kern + k * (CIN * COUT);
    #pragma unroll
    for (int it = 0; it < (CIN * COUT) / 256; ++it) {
        int idx = tid + it * 256;
        int c = idx >> 6;
        int o = idx & 63;
        lds_b[(c >> 1) * 128 + (o << 1) + (c & 1)] = kb[idx];
    }
    __syncthreads();

    const int wave = tid >> 5;
    const int lane = tid & 31;
    const int half = lane >> 4;     // 0: lanes 0-15 (slots K0/K1), 1: lanes 16-31 (K2/K3)
    const int lm   = lane & 15;

    const int tile = (blockIdx.x % BLOCKS_PER_K) * WAVES_PER_BLOCK + wave;
    const int base = tile * 16;     // pair base within this k

    const int* im = in_map  + k * PAIRS + base;
    const int* om = out_map + k * PAIRS + base;

    // A gather: lane's pair row, 16B-aligned float4 chunks at col 8t + 4*half
    const int    in_row = im[lm];
    const float* arow   = feats + (long)in_row * CIN + half * 4;

    v8f acc0 = {}, acc1 = {}, acc2 = {}, acc3 = {};
    const v2f* bl = (const v2f*)lds_b;

    #pragma unroll
    for (int t = 0; t < 8; ++t) {
        v4f a4 = *(const v4f*)(arow + 8 * t);
        v2f a0 = {a4.x, a4.y};                       // step 2t fragment
        v2f a1 = {a4.z, a4.w};                       // step 2t+1 fragment

        const int bi0 = (4 * t +     2 * half) * 64 + lm;   // step 2t B pairs
        const int bi1 = (4 * t + 1 + 2 * half) * 64 + lm;   // step 2t+1 B pairs

        v2f b00 = bl[bi0 +  0], b01 = bl[bi0 + 16], b02 = bl[bi0 + 32], b03 = bl[bi0 + 48];
        v2f b10 = bl[bi1 +  0], b11 = bl[bi1 + 16], b12 = bl[bi1 + 32], b13 = bl[bi1 + 48];

        acc0 = __builtin_amdgcn_wmma_f32_16x16x4_f32(false, a0, false, b00, (short)0, acc0, false, false);
        acc1 = __builtin_amdgcn_wmma_f32_16x16x4_f32(false, a0, false, b01, (short)0, acc1, false, false);
        acc2 = __builtin_amdgcn_wmma_f32_16x16x4_f32(false, a0, false, b02, (short)0, acc2, false, false);
        acc3 = __builtin_amdgcn_wmma_f32_16x16x4_f32(false, a0, false, b03, (short)0, acc3, false, false);

        acc0 = __builtin_amdgcn_wmma_f32_16x16x4_f32(false, a1, false, b10, (short)0, acc0, false, false);
        acc1 = __builtin_amdgcn_wmma_f32_16x16x4_f32(false, a1, false, b11, (short)0, acc1, false, false);
        acc2 = __builtin_amdgcn_wmma_f32_16x16x4_f32(false, a1, false, b12, (short)0, acc2, false, false);
        acc3 = __builtin_amdgcn_wmma_f32_16x16x4_f32(false, a1, false, b13, (short)0, acc3, false, false);
    }

    // Scatter-add: VGPR j of acc holds row M = half*8 + j, column lm (+16*otile)
    #pragma unroll
    for (int j = 0; j < 8; ++j) {
        const int M    = half * 8 + j;
        const int orow = om[M];
        float* op = out + (long)orow * COUT + lm;
        unsafeAtomicAdd(op +  0, acc0[j]);
        unsafeAtomicAdd(op + 16, acc1[j]);
        unsafeAtomicAdd(op + 32, acc2[j]);
        unsafeAtomicAdd(op + 48, acc3[j]);
    }
}

// ---------------------------------------------------------------------------
extern "C" void kernel_launch(void* const* d_in, const int* in_sizes, int n_in,
                              void* d_out, int out_size, void* d_ws, size_t ws_size,
                              hipStream_t stream) {
    const float* feats   = (const float*)d_in[0];  // [N_IN, 64]
    const float* kern    = (const float*)d_in[1];  // [27, 64, 64]
    const float* bias    = (const float*)d_in[2];  // [1, 64]
    const int*   in_map  = (const int*)  d_in[3];  // [27, 65536]
    const int*   out_map = (const int*)  d_in[4];  // [27, 65536]
    float*       out     = (float*)d_out;          // [N_OUT, 64]

    // 1) out = bias broadcast
    sconv_init_bias<<<(N_OUT_V * COUT) / 256, 256, 0, stream>>>(bias, out);

    // 2) gather -> WMMA -> atomic scatter-add
    sconv_wmma_f32<<<KVOL * BLOCKS_PER_K, 256, 0, stream>>>(feats, kern, in_map, out_map, out);
}